// CMValidator_10986526343848
// MI455X (gfx1250) — compile-verified
//
#include <hip/hip_runtime.h>
#include <stdint.h>

typedef float v2f __attribute__((ext_vector_type(2)));
typedef float v8f __attribute__((ext_vector_type(8)));

#define WAVES_PER_BLOCK 4
#define ROW_STRIDE 132                 // dwords per staged row: 128 data + 4 pad (kills LDS bank conflicts)
#define WAVE_LDS_DW (16 * ROW_STRIDE)  // 2112 dwords per wave
#define G_STRIDE 17                    // 16x16 Gram tile row stride (pad 1)
#define WAVE_G_DW (16 * G_STRIDE)

// Each wave: 3 batches. R = 16 x 128 tile (rows 0..14 = 3 batches x 5 verts, row 15 unused).
// G = R * R^T via 32x V_WMMA_F32_16X16X4_F32, A and B share the same VGPRs (symmetric Gram).
__global__ void __launch_bounds__(WAVES_PER_BLOCK * 32)
cm_validator_kernel(const float* __restrict__ verts, float* __restrict__ out, int B)
{
    __shared__ float smem[WAVES_PER_BLOCK * WAVE_LDS_DW]; // 33792 B
    __shared__ float gsm [WAVES_PER_BLOCK * WAVE_G_DW];   //  4352 B

    const int  lane = threadIdx.x & 31;
    const int  wib  = threadIdx.x >> 5;
    const long wave = (long)blockIdx.x * WAVES_PER_BLOCK + wib;
    const long batchBase = wave * 3;

    float* wsm = &smem[wib * WAVE_LDS_DW];
    float* wg  = &gsm [wib * WAVE_G_DW];

    // ---- Stage 15 rows (3 batches x 5 verts x 128 f32) global -> LDS.
    // One async instruction per row: 32 lanes x 16B = 512B, perfectly coalesced.
    #pragma unroll
    for (int r = 0; r < 15; ++r) {
        int  q   = r / 5;
        int  vtx = r - 5 * q;
        long b   = batchBase + q; if (b > (long)B - 1) b = (long)B - 1;  // clamp (no divergence)
        const float* gp = verts + ((b * 5 + vtx) << 7) + (lane << 2);
        unsigned lds = (unsigned)(size_t)(wsm + r * ROW_STRIDE + (lane << 2));
        unsigned long long ga = (unsigned long long)(size_t)gp;
        asm volatile("global_load_async_to_lds_b128 %0, %1, off"
                     :: "v"(lds), "v"(ga) : "memory");
    }
    asm volatile("s_wait_asynccnt 0x0" ::: "memory");

    // ---- Gram via WMMA. A (16x4 f32) layout: lanes 0-15 hold M=lane, K=0,1; lanes 16-31 hold
    // M=lane-16, K=2,3. B (4x16) mirrors with N=lane -> identical register image for R*R^T.
    const int half = lane >> 4;
    const int rsel = lane & 15;
    const float* abase = wsm + rsel * ROW_STRIDE + half * 2;

    v8f acc0 = {}, acc1 = {}, acc2 = {}, acc3 = {};
    #pragma unroll
    for (int c = 0; c < 32; c += 4) {
        v2f a0 = *(const v2f*)(abase + (c + 0) * 4);
        v2f a1 = *(const v2f*)(abase + (c + 1) * 4);
        v2f a2 = *(const v2f*)(abase + (c + 2) * 4);
        v2f a3 = *(const v2f*)(abase + (c + 3) * 4);
        acc0 = __builtin_amdgcn_wmma_f32_16x16x4_f32(false, a0, false, a0, (short)0, acc0, false, false);
        acc1 = __builtin_amdgcn_wmma_f32_16x16x4_f32(false, a1, false, a1, (short)0, acc1, false, false);
        acc2 = __builtin_amdgcn_wmma_f32_16x16x4_f32(false, a2, false, a2, (short)0, acc2, false, false);
        acc3 = __builtin_amdgcn_wmma_f32_16x16x4_f32(false, a3, false, a3, (short)0, acc3, false, false);
    }
    v8f g = (acc0 + acc1) + (acc2 + acc3);

    // ---- Spill 16x16 Gram tile to LDS (C layout: lane = N (mod 16), vgpr v -> M = v + 8*(lane/16)).
    #pragma unroll
    for (int v = 0; v < 8; ++v)
        wg[(v + half * 8) * G_STRIDE + rsel] = g[v];
    asm volatile("" ::: "memory"); // DS ops are in-order within a wave; just stop compiler reordering

    // ---- d2 pair outputs: 3 batches x 10 pairs -> lanes 0..29.
    if (lane < 30) {
        int  q = lane / 10;
        int  p = lane - q * 10;
        long b = batchBase + q;
        if (b < (long)B) {
            int i  = (int)((0x3221110000ull >> (4 * p)) & 0xF); // PI
            int j  = (int)((0x4434324321ull >> (4 * p)) & 0xF); // PJ
            int r0 = 5 * q + i, c0 = 5 * q + j;
            float gii = wg[r0 * G_STRIDE + r0];
            float gjj = wg[c0 * G_STRIDE + c0];
            float gij = wg[r0 * G_STRIDE + c0];
            float d2  = fmaxf(gii + gjj - 2.0f * gij, 0.0f);
            out[b * 10 + p] = d2;
        }
    }

    // ---- Cayley-Menger det (6x6, LU w/ compare-swap partial pivoting): lanes 0..2, one batch each.
    if (lane < 3) {
        long b = batchBase + lane;
        if (b < (long)B) {
            const int qb = 5 * lane;
            float nrm[5];
            #pragma unroll
            for (int i = 0; i < 5; ++i) nrm[i] = wg[(qb + i) * G_STRIDE + (qb + i)];

            float m[6][6];
            m[0][0] = 0.0f;
            #pragma unroll
            for (int i = 1; i < 6; ++i) { m[0][i] = 1.0f; m[i][0] = 1.0f; }
            #pragma unroll
            for (int i = 0; i < 5; ++i) {
                #pragma unroll
                for (int j = 0; j < 5; ++j) {
                    float gij = wg[(qb + i) * G_STRIDE + (qb + j)];
                    m[i + 1][j + 1] = fmaxf(nrm[i] + nrm[j] - 2.0f * gij, 0.0f);
                }
            }

            float det = 1.0f;
            #pragma unroll
            for (int c = 0; c < 6; ++c) {
                #pragma unroll
                for (int r = c + 1; r < 6; ++r) {          // predicated pivot swaps, static indexing
                    bool sw = fabsf(m[r][c]) > fabsf(m[c][c]);
                    #pragma unroll
                    for (int k = c; k < 6; ++k) {
                        float t0 = m[c][k], t1 = m[r][k];
                        m[c][k] = sw ? t1 : t0;
                        m[r][k] = sw ? t0 : t1;
                    }
                    det = sw ? -det : det;
                }
                float piv = m[c][c];
                det *= piv;
                float inv = (piv != 0.0f) ? (1.0f / piv) : 0.0f;
                #pragma unroll
                for (int r = c + 1; r < 6; ++r) {
                    float f = m[r][c] * inv;
                    #pragma unroll
                    for (int k = c + 1; k < 6; ++k)
                        m[r][k] -= f * m[c][k];
                }
            }
            out[(size_t)B * 10 + b] = (-1.0f / 9216.0f) * det; // PREFACTOR for K=4
        }
    }
}

extern "C" void kernel_launch(void* const* d_in, const int* in_sizes, int n_in,
                              void* d_out, int out_size, void* d_ws, size_t ws_size,
                              hipStream_t stream) {
    const float* verts = (const float*)d_in[0];
    float* out = (float*)d_out;
    const int B = in_sizes[0] / (5 * 128);           // 262144
    const int batchesPerBlock = WAVES_PER_BLOCK * 3; // 12
    const int blocks = (B + batchesPerBlock - 1) / batchesPerBlock;
    cm_validator_kernel<<<blocks, WAVES_PER_BLOCK * 32, 0, stream>>>(verts, out, B);
}